// IPCTKContact_21869973471370
// MI455X (gfx1250) — compile-verified
//
#include <hip/hip_runtime.h>

// ---------- problem constants (match reference) ----------
#define N_VERTS      262144
#define N_PAIRS      8388608
#define THREADS      256
#define PAIRS_PER_THREAD 16            // 4 x int4 per thread
#define PAIRS_PER_BLOCK  (THREADS * PAIRS_PER_THREAD)   // 4096
#define NBLK_PAIR    (N_PAIRS / PAIRS_PER_BLOCK)        // 2048
#define DHAT2        0.0025f           // (0.05)^2
#define INV_DHAT2    400.0f
#define EPSF         1e-12f

typedef __attribute__((ext_vector_type(2)))  float v2f;
typedef __attribute__((ext_vector_type(4)))  float v4f;
typedef __attribute__((ext_vector_type(8)))  float v8f;
typedef __attribute__((ext_vector_type(4)))  int   v4i;

// ---------------------------------------------------------------
// Kernel 1: coords = rest_positions + Uu  (flat f32 add, x4 vectorized)
// coords lives in d_ws and stays L2-resident for the gather phase.
// ---------------------------------------------------------------
__global__ void coords_kernel(const float* __restrict__ rest,
                              const float* __restrict__ Uu,
                              float* __restrict__ coords, int n4) {
  int i = blockIdx.x * blockDim.x + threadIdx.x;
  if (i < n4) {
    v4f r = __builtin_nontemporal_load((const v4f*)rest + i);
    v4f u = __builtin_nontemporal_load((const v4f*)Uu   + i);
    *((v4f*)coords + i) = r + u;   // regular store: keep hot in L2
  }
}

// ---------------------------------------------------------------
// Kernel 2: per-pair barrier term + deterministic per-block reduction.
// Index streams read once -> non-temporal coalesced B128 loads (96 MB @
// 23.3 TB/s is the roofline). Coord gathers hit the 2MB L2-resident table.
// Reduction: 5 wave-level shfl adds, then one LDS pass (fixed order).
// ---------------------------------------------------------------
__global__ void __launch_bounds__(THREADS)
pair_kernel(const int* __restrict__ pv,
            const int* __restrict__ pe0,
            const int* __restrict__ pe1,
            const float* __restrict__ coords,
            float* __restrict__ partials) {
  const v4i* __restrict__ pv4  = (const v4i*)pv;
  const v4i* __restrict__ pe04 = (const v4i*)pe0;
  const v4i* __restrict__ pe14 = (const v4i*)pe1;
  const v2f* __restrict__ cpos = (const v2f*)coords;

  const int t = threadIdx.x;
  const int base = blockIdx.x * (PAIRS_PER_BLOCK / 4);   // int4 units

  float acc = 0.0f;

#pragma unroll
  for (int it = 0; it < PAIRS_PER_THREAD / 4; ++it) {
    const int vi = base + it * THREADS + t;              // coalesced B128
    v4i v  = __builtin_nontemporal_load(pv4  + vi);
    v4i e0 = __builtin_nontemporal_load(pe04 + vi);
    v4i e1 = __builtin_nontemporal_load(pe14 + vi);

#pragma unroll
    for (int j = 0; j < 4; ++j) {
      v2f p = cpos[v[j]];
      v2f a = cpos[e0[j]];
      v2f b = cpos[e1[j]];

      float abx = b.x - a.x, aby = b.y - a.y;
      float apx = p.x - a.x, apy = p.y - a.y;

      float denom = __builtin_fmaf(abx, abx, aby * aby);
      float tnum  = __builtin_fmaf(apx, abx, apy * aby);
      float tt    = tnum / fmaxf(denom, EPSF);
      tt = fminf(fmaxf(tt, 0.0f), 1.0f);

      float dx = __builtin_fmaf(-tt, abx, apx);
      float dy = __builtin_fmaf(-tt, aby, apy);
      float d2 = __builtin_fmaf(dx, dx, dy * dy);

      float d2s  = fmaxf(d2, EPSF);
      float g    = d2s - DHAT2;
      float term = -(g * g) * logf(d2s * INV_DHAT2);
      acc += (d2 < DHAT2) ? term : 0.0f;
    }
  }

  // wave32 reduction: 5 shfl steps, no barriers
#pragma unroll
  for (int off = 16; off > 0; off >>= 1)
    acc += __shfl_down(acc, off, 32);

  __shared__ float swave[THREADS / 32];
  if ((t & 31) == 0) swave[t >> 5] = acc;
  __syncthreads();
  if (t == 0) {
    float s = 0.0f;
#pragma unroll
    for (int w = 0; w < THREADS / 32; ++w) s += swave[w];   // fixed order
    partials[blockIdx.x] = s;
  }
}

// ---------------------------------------------------------------
// Kernel 3: final reduction of 2048 partials.
// Stage 1: 256 threads, two independent B128 loads each (one L2 round
// trip, no serialized-load loop), LDS to 256 values.
// Stage 2: wave 0 (uniform branch -> EXEC all-1s) folds 256 -> 32x2 and
// collapses across lanes with one exact-f32 V_WMMA_F32_16X16X4_F32
// (A = lane partials, B = ones => D[m][n] = sum_k A[m][k]), then sums the
// 8 D registers + shfl_down(16) for rows 8..15 living in lanes 16..31.
// ---------------------------------------------------------------
__global__ void __launch_bounds__(256)
final_reduce(const float* __restrict__ partials, float* __restrict__ out) {
  const int t = threadIdx.x;
  const v4f* __restrict__ p4 = (const v4f*)partials;   // 512 x v4f

  v4f x = p4[t];
  v4f y = p4[t + 256];
  float s = ((x.x + x.y) + (x.z + x.w)) + ((y.x + y.y) + (y.z + y.w));

  __shared__ float sdata[256];
  sdata[t] = s;
  __syncthreads();

  if (t < 32) {                          // whole wave 0: EXEC all ones
    float s0 = (sdata[t]      + sdata[t + 64])  + (sdata[t + 128] + sdata[t + 192]);
    float s1 = (sdata[t + 32] + sdata[t + 96])  + (sdata[t + 160] + sdata[t + 224]);

    v2f a; a.x = s0;   a.y = s1;
    v2f b; b.x = 1.0f; b.y = 1.0f;
    v8f c = {};
    c = __builtin_amdgcn_wmma_f32_16x16x4_f32(
        /*neg_a=*/false, a, /*neg_b=*/false, b,
        /*c_mod=*/(short)0, c, /*reuse_a=*/false, /*reuse_b=*/false);

    float r = ((c[0] + c[1]) + (c[2] + c[3])) + ((c[4] + c[5]) + (c[6] + c[7]));
    r += __shfl_down(r, 16, 32);
    if (t == 0) out[0] = r;
  }
}

// ---------------------------------------------------------------
extern "C" void kernel_launch(void* const* d_in, const int* in_sizes, int n_in,
                              void* d_out, int out_size, void* d_ws, size_t ws_size,
                              hipStream_t stream) {
  const float* Uu   = (const float*)d_in[0];
  const float* rest = (const float*)d_in[1];
  const int*   pv   = (const int*)d_in[2];
  const int*   pe0  = (const int*)d_in[3];
  const int*   pe1  = (const int*)d_in[4];
  float* out = (float*)d_out;

  float* coords   = (float*)d_ws;                               // 2 MB
  float* partials = (float*)((char*)d_ws + (size_t)N_VERTS * 2 * sizeof(float));

  const int n4 = (N_VERTS * 2) / 4;                             // 131072
  coords_kernel<<<(n4 + THREADS - 1) / THREADS, THREADS, 0, stream>>>(
      rest, Uu, coords, n4);

  pair_kernel<<<NBLK_PAIR, THREADS, 0, stream>>>(pv, pe0, pe1, coords, partials);

  final_reduce<<<1, 256, 0, stream>>>(partials, out);
}